// CRF_59167469470117
// MI455X (gfx1250) — compile-verified
//
#include <hip/hip_runtime.h>
#include <hip/hip_bf16.h>
#include <math.h>

typedef __attribute__((ext_vector_type(2))) float v2f;
typedef __attribute__((ext_vector_type(8))) float v8f;

#define S_DIM 512
#define B_DIM 256
#define D_DIM 512
#define L_DIM 50
#define LPAD  64
#define KCH   (D_DIM / 4)   // 128 k-chunks of 4

// ---------------------------------------------------------------------------
// Kernel 1: emissions[r][l] = sum_d feats[r][d] * W[d][l] + b[l]
//   r = s*B + b, output padded to LPAD=64 columns.
//   fp32 WMMA 16x16x4. W is pre-swizzled into LDS in B-fragment layout:
//   Wfrag[kc][ntile][lane] = (W[kc*4+koff][n], W[kc*4+koff+1][n]),
//   koff = 2*(lane/16), n = ntile*16 + lane%16.  (128 KB LDS)
// ---------------------------------------------------------------------------
__global__ __launch_bounds__(256) void crf_emissions_wmma(
    const float* __restrict__ feats, const float* __restrict__ W,
    const float* __restrict__ bias, float* __restrict__ emis, int rows)
{
  __shared__ v2f   Wfrag[KCH * 4 * 32];   // 128 KB
  __shared__ float blds[LPAD];

  const int tid = threadIdx.x;

  // Cooperative swizzle of W (D x L, row-major) into WMMA B-fragment order.
  for (int idx = tid; idx < KCH * 4 * 32; idx += 256) {
    const int lane = idx & 31;
    const int nt   = (idx >> 5) & 3;
    const int kc   = idx >> 7;
    const int koff = (lane >> 4) << 1;
    const int n    = nt * 16 + (lane & 15);
    const int k    = kc * 4 + koff;
    v2f w; w.x = 0.0f; w.y = 0.0f;
    if (n < L_DIM) {
      w.x = W[(size_t)k * L_DIM + n];
      w.y = W[(size_t)(k + 1) * L_DIM + n];
    }
    Wfrag[idx] = w;
  }
  if (tid < LPAD) blds[tid] = (tid < L_DIM) ? bias[tid] : 0.0f;
  __syncthreads();

  const int wave = tid >> 5;
  const int lane = tid & 31;
  const int koff = (lane >> 4) << 1;   // A-frag K offset for this lane half
  const int mtiles = rows >> 4;        // 8192 tiles of 16 rows

  for (int tile = blockIdx.x * 8 + wave; tile < mtiles; tile += gridDim.x * 8) {
    const float* arow =
        feats + (size_t)(tile * 16 + (lane & 15)) * D_DIM + koff;
    v8f c0 = {}; v8f c1 = {}; v8f c2 = {}; v8f c3 = {};
    for (int kc = 0; kc < KCH; ++kc) {
      const v2f a  = *(const v2f*)(arow + kc * 4);
      const v2f* bf = &Wfrag[kc * 128];
      const v2f b0 = bf[lane];
      const v2f b1 = bf[32 + lane];
      const v2f b2 = bf[64 + lane];
      const v2f b3 = bf[96 + lane];
      c0 = __builtin_amdgcn_wmma_f32_16x16x4_f32(false, a, false, b0, (short)0, c0, false, false);
      c1 = __builtin_amdgcn_wmma_f32_16x16x4_f32(false, a, false, b1, (short)0, c1, false, false);
      c2 = __builtin_amdgcn_wmma_f32_16x16x4_f32(false, a, false, b2, (short)0, c2, false, false);
      c3 = __builtin_amdgcn_wmma_f32_16x16x4_f32(false, a, false, b3, (short)0, c3, false, false);
    }
    // C layout: VGPR v -> row (v + 8*(lane/16)) of tile, col = lane%16
    const int n0    = lane & 15;
    const int rbase = tile * 16 + ((lane >> 4) << 3);
    float* out = emis + (size_t)rbase * LPAD + n0;
#pragma unroll
    for (int v = 0; v < 8; ++v) {
      out[(size_t)v * LPAD +  0] = c0[v] + blds[n0];
      out[(size_t)v * LPAD + 16] = c1[v] + blds[n0 + 16];
      out[(size_t)v * LPAD + 32] = c2[v] + blds[n0 + 32];
      out[(size_t)v * LPAD + 48] = c3[v] + blds[n0 + 48];
    }
  }
}

// ---------------------------------------------------------------------------
// Kernel 2: per-batch forward scan + gold score.  One 64-thread WG per batch.
// ---------------------------------------------------------------------------
__global__ __launch_bounds__(64) void crf_scan_score(
    const float* __restrict__ emis, const int* __restrict__ tags,
    const int* __restrict__ lens, const float* __restrict__ start,
    const float* __restrict__ trans, const float* __restrict__ endt,
    float* __restrict__ partial)
{
  const int b   = blockIdx.x;
  const int tid = threadIdx.x;

  __shared__ float tlds[L_DIM * 65];   // trans, row stride 65 (bank-friendly)
  __shared__ float alpha[64];
  __shared__ float red[64];

  for (int idx = tid; idx < L_DIM * L_DIM; idx += 64) {
    const int j = idx / L_DIM;
    const int i = idx - j * L_DIM;
    tlds[j * 65 + i] = trans[idx];
  }
  const int len = lens[b];
  if (tid < L_DIM)
    alpha[tid] = emis[(size_t)b * LPAD + tid] + start[tid];
  __syncthreads();

  for (int t = 1; t < S_DIM; ++t) {
    float nv = 0.0f;
    if (tid < L_DIM) {
      const float* tr = &tlds[tid * 65];
      float m = -INFINITY;
      for (int i = 0; i < L_DIM; ++i) m = fmaxf(m, alpha[i] + tr[i]);
      float s = 0.0f;
      for (int i = 0; i < L_DIM; ++i) s += __expf(alpha[i] + tr[i] - m);
      nv = m + __logf(s) + emis[((size_t)t * B_DIM + b) * LPAD + tid];
    }
    __syncthreads();
    if (tid < L_DIM && t < len) alpha[tid] = nv;
    __syncthreads();
  }

  // forward_score = logsumexp_j(alpha[j] + end[j])
  const float v = (tid < L_DIM) ? (alpha[tid] + endt[tid]) : -INFINITY;
  red[tid] = v;
  __syncthreads();
  for (int off = 32; off > 0; off >>= 1) {
    if (tid < off) red[tid] = fmaxf(red[tid], red[tid + off]);
    __syncthreads();
  }
  const float m = red[0];
  __syncthreads();
  red[tid] = (tid < L_DIM) ? __expf(v - m) : 0.0f;
  __syncthreads();
  for (int off = 32; off > 0; off >>= 1) {
    if (tid < off) red[tid] += red[tid + off];
    __syncthreads();
  }
  const float fwd = m + __logf(red[0]);
  __syncthreads();

  // gold score
  float g = 0.0f;
  for (int t = 1 + tid; t < S_DIM; t += 64) {
    if (t < len) {
      const int tt = tags[t * B_DIM + b];
      const int tp = tags[(t - 1) * B_DIM + b];
      g += tlds[tt * 65 + tp] + emis[((size_t)t * B_DIM + b) * LPAD + tt];
    }
  }
  if (tid == 0) {
    const int t0 = tags[b];
    g += start[t0] + emis[(size_t)b * LPAD + t0];
    g += endt[tags[(len - 1) * B_DIM + b]];
  }
  red[tid] = g;
  __syncthreads();
  for (int off = 32; off > 0; off >>= 1) {
    if (tid < off) red[tid] += red[tid + off];
    __syncthreads();
  }
  if (tid == 0) partial[b] = fwd - red[0];
}

// ---------------------------------------------------------------------------
// Kernel 3: deterministic sum of 256 per-batch scores.
// ---------------------------------------------------------------------------
__global__ __launch_bounds__(256) void crf_final_reduce(
    const float* __restrict__ partial, float* __restrict__ out)
{
  __shared__ float red[256];
  const int tid = threadIdx.x;
  red[tid] = partial[tid];
  __syncthreads();
  for (int off = 128; off > 0; off >>= 1) {
    if (tid < off) red[tid] += red[tid + off];
    __syncthreads();
  }
  if (tid == 0) out[0] = red[0];
}

extern "C" void kernel_launch(void* const* d_in, const int* in_sizes, int n_in,
                              void* d_out, int out_size, void* d_ws, size_t ws_size,
                              hipStream_t stream) {
  (void)in_sizes; (void)n_in; (void)out_size; (void)ws_size;
  const float* feats = (const float*)d_in[0];
  const int*   tags  = (const int*)d_in[1];
  const int*   lens  = (const int*)d_in[2];
  const float* W     = (const float*)d_in[3];
  const float* bias  = (const float*)d_in[4];
  const float* start = (const float*)d_in[5];
  const float* trans = (const float*)d_in[6];
  const float* endt  = (const float*)d_in[7];

  float* emis    = (float*)d_ws;                                   // S*B*64 fp32
  float* partial = (float*)((char*)d_ws +
                   (size_t)S_DIM * B_DIM * LPAD * sizeof(float));  // B fp32

  const int rows = S_DIM * B_DIM;
  crf_emissions_wmma<<<1024, 256, 0, stream>>>(feats, W, bias, emis, rows);
  crf_scan_score<<<B_DIM, 64, 0, stream>>>(emis, tags, lens, start, trans,
                                           endt, partial);
  crf_final_reduce<<<1, 256, 0, stream>>>(partial, (float*)d_out);
}